// SinkhornEMDLoss_40699110097304
// MI455X (gfx1250) — compile-verified
//
#include <hip/hip_runtime.h>
#include <hip/hip_bf16.h>

// Sinkhorn EMD loss, MI455X (gfx1250, wave32).
// B=4, N=1024, REG=0.1, 50 iterations. Output: scalar f32.
//
// Strategy: cost matrix (16MB, L2-resident) built once with
// V_WMMA_F32_16X16X4_F32 (K=3 padded to 4, fp32-exact). Sinkhorn kept in the
// u/v potential form: logP = -cost/REG - u[n] - v[m], so each half-iteration
// only READS cost (L2) and writes a 16KB potential vector. Row/col passes are
// logsumexp reductions, vectorized b128 + segmented for L2-latency hiding.

#define BATCH 4
#define NPTS 1024
#define INV_REG 10.0f       // 1/0.1
#define ITERS 50

typedef __attribute__((ext_vector_type(2))) float v2f;
typedef __attribute__((ext_vector_type(8))) float v8f;

// ---------------------------------------------------------------------------
// 0) zero the column potentials (v). u is fully overwritten by the first row
//    pass, so only v needs a clean start each launch (harness never re-poisons).
__global__ void sk_init(float* __restrict__ vpot) {
    int i = blockIdx.x * blockDim.x + threadIdx.x;
    if (i < BATCH * NPTS) vpot[i] = 0.0f;
}

// ---------------------------------------------------------------------------
// 1) cost[b,n,m] = relu(|x_n|^2 + |y_m|^2 - 2 x_n.y_m), xy tile via
//    V_WMMA_F32_16X16X4_F32. One wave per 16x16 tile, 8 waves per block.
__global__ void sk_cost_wmma(const float* __restrict__ pred,
                             const float* __restrict__ gt,
                             float* __restrict__ cost) {
    const int lane   = threadIdx.x & 31;
    const int waveId = threadIdx.x >> 5;
    const int tile   = blockIdx.x * 8 + waveId;     // 16384 tiles total
    const int b      = tile >> 12;                  // tile / (64*64)
    const int rem    = tile & 4095;
    const int n0     = (rem >> 6) << 4;             // row tile * 16
    const int m0     = (rem & 63) << 4;             // col tile * 16

    const int hf = lane >> 4;                       // half-wave: 0 or 1
    const int lr = lane & 15;

    // A fragment (16x4 f32): lanes 0-15 hold K=0,1 ; lanes 16-31 hold K=2,3(=0 pad)
    const float* pr = pred + ((size_t)b * NPTS + n0 + lr) * 3;
    v2f a;
    a.x = hf ? pr[2] : pr[0];
    a.y = hf ? 0.0f  : pr[1];

    // B fragment (4x16 f32): same K striping, N across lanes
    const float* gr = gt + ((size_t)b * NPTS + m0 + lr) * 3;
    v2f bm;
    bm.x = hf ? gr[2] : gr[0];
    bm.y = hf ? 0.0f  : gr[1];

    v8f c = {};
    c = __builtin_amdgcn_wmma_f32_16x16x4_f32(
            false, a, false, bm, (short)0, c, false, false);

    const float g0 = gr[0], g1 = gr[1], g2 = gr[2];
    const float y2 = g0 * g0 + g1 * g1 + g2 * g2;

    // C/D layout: vgpr v, lane L -> (M = v + (L>=16)*8, N = L&15)
    #pragma unroll
    for (int v = 0; v < 8; ++v) {
        const int row = n0 + v + hf * 8;
        const float* p = pred + ((size_t)b * NPTS + row) * 3;
        const float x2 = p[0] * p[0] + p[1] * p[1] + p[2] * p[2];
        float cval = x2 + y2 - 2.0f * c[v];
        cval = fmaxf(cval, 0.0f);
        cost[((size_t)b * NPTS + row) * NPTS + (m0 + lr)] = cval;
    }
}

// ---------------------------------------------------------------------------
// 2a) row pass: u[b,n] = LSE_m( -cost[b,n,m]/REG - v[m] )
//     One wave per row; 8x b128 loads per lane (32 elems in VGPRs); butterfly.
__global__ void sk_row_lse(const float* __restrict__ cost,
                           const float* __restrict__ vpot,
                           float* __restrict__ upot) {
    const int lane   = threadIdx.x & 31;
    const int waveId = threadIdx.x >> 5;
    const int row    = blockIdx.x * 8 + waveId;     // 0 .. 4095
    const int b      = row >> 10;

    const float4* crow = (const float4*)(cost + (size_t)row * NPTS);
    const float4* vrow = (const float4*)(vpot + b * NPTS);

    float t[32];
    #pragma unroll
    for (int k = 0; k < 8; ++k) {
        const int q = lane + (k << 5);              // float4 index; 512B/step wave-wide
        const float4 cv = crow[q];
        const float4 vv = vrow[q];
        t[4 * k + 0] = -cv.x * INV_REG - vv.x;
        t[4 * k + 1] = -cv.y * INV_REG - vv.y;
        t[4 * k + 2] = -cv.z * INV_REG - vv.z;
        t[4 * k + 3] = -cv.w * INV_REG - vv.w;
    }

    float mx = t[0];
    #pragma unroll
    for (int k = 1; k < 32; ++k) mx = fmaxf(mx, t[k]);
    #pragma unroll
    for (int off = 16; off >= 1; off >>= 1)
        mx = fmaxf(mx, __shfl_xor(mx, off, 32));

    float s = 0.0f;
    #pragma unroll
    for (int k = 0; k < 32; ++k) s += __expf(t[k] - mx);
    #pragma unroll
    for (int off = 16; off >= 1; off >>= 1)
        s += __shfl_xor(s, off, 32);

    if (lane == 0) upot[row] = mx + __logf(s);
}

// 2b) col pass: v[b,m] = LSE_n( -cost[b,n,m]/REG - u[n] )
//     Block = 256 threads: lanes-of-32 own 4 adjacent columns each via b128
//     loads (128 cols/block), 8 row-segments of 128 rows run in parallel and
//     merge their online-LSE states through LDS (exact log-domain combine).
__global__ void sk_col_lse(const float* __restrict__ cost,
                           const float* __restrict__ upot,
                           float* __restrict__ vpot) {
    __shared__ float smx[8][128];
    __shared__ float ssm[8][128];

    const int lane = threadIdx.x & 31;
    const int seg  = threadIdx.x >> 5;              // 0..7 -> rows seg*128..+127
    const int gcol = blockIdx.x * 128 + lane * 4;   // flat col in [0, B*N)
    const int b    = gcol >> 10;
    const int m    = gcol & (NPTS - 1);

    const float* cc = cost + (size_t)b * NPTS * NPTS + m;
    const float* ur = upot + b * NPTS;

    float mx0 = -__builtin_inff(), mx1 = mx0, mx2 = mx0, mx3 = mx0;
    float s0 = 0.0f, s1 = 0.0f, s2 = 0.0f, s3 = 0.0f;

    const int nBeg = seg * 128, nEnd = nBeg + 128;
    #pragma unroll 4
    for (int n = nBeg; n < nEnd; ++n) {
        if (n + 8 < nEnd)
            __builtin_prefetch(cc + (size_t)(n + 8) * NPTS, 0, 0);
        const float4 cv = *(const float4*)(cc + (size_t)n * NPTS);
        const float un = ur[n];                      // scalar across the wave
        float x, nm;
        x = -cv.x * INV_REG - un; nm = fmaxf(mx0, x);
        s0 = s0 * __expf(mx0 - nm) + __expf(x - nm); mx0 = nm;
        x = -cv.y * INV_REG - un; nm = fmaxf(mx1, x);
        s1 = s1 * __expf(mx1 - nm) + __expf(x - nm); mx1 = nm;
        x = -cv.z * INV_REG - un; nm = fmaxf(mx2, x);
        s2 = s2 * __expf(mx2 - nm) + __expf(x - nm); mx2 = nm;
        x = -cv.w * INV_REG - un; nm = fmaxf(mx3, x);
        s3 = s3 * __expf(mx3 - nm) + __expf(x - nm); mx3 = nm;
    }

    const int c0 = lane * 4;
    smx[seg][c0 + 0] = mx0; ssm[seg][c0 + 0] = s0;
    smx[seg][c0 + 1] = mx1; ssm[seg][c0 + 1] = s1;
    smx[seg][c0 + 2] = mx2; ssm[seg][c0 + 2] = s2;
    smx[seg][c0 + 3] = mx3; ssm[seg][c0 + 3] = s3;
    __syncthreads();

    // 128 combiner threads: merge 8 segment states per column (fixed order).
    const int tid = threadIdx.x;
    if (tid < 128) {
        float M = smx[0][tid];
        #pragma unroll
        for (int sg = 1; sg < 8; ++sg) M = fmaxf(M, smx[sg][tid]);
        float S = 0.0f;
        #pragma unroll
        for (int sg = 0; sg < 8; ++sg)
            S += ssm[sg][tid] * __expf(smx[sg][tid] - M);
        vpot[blockIdx.x * 128 + tid] = M + __logf(S);
    }
}

// ---------------------------------------------------------------------------
// 3) loss partials: sum exp(-cost/REG - u - v) * cost, b128 loads,
//    fixed-order LDS tree reduce per block (deterministic).
__global__ void sk_final_partial(const float* __restrict__ cost,
                                 const float* __restrict__ upot,
                                 const float* __restrict__ vpot,
                                 float* __restrict__ partials) {
    __shared__ float sdata[256];
    const int tid    = threadIdx.x;
    const int base   = blockIdx.x * blockDim.x + tid;      // float4 index
    const int total4 = (BATCH * NPTS * NPTS) >> 2;         // 1M float4s
    const int stride = gridDim.x * blockDim.x;             // 256K -> 4 iters

    float acc = 0.0f;
    for (int i = base; i < total4; i += stride) {
        const int e  = i << 2;                              // element index
        const int b  = e >> 20;
        const int r  = e & ((NPTS * NPTS) - 1);
        const int n  = r >> 10;
        const int mb = r & (NPTS - 1);                      // multiple of 4
        const float4 cv = *(const float4*)(cost + (size_t)e);
        const float4 vv = *(const float4*)(vpot + b * NPTS + mb);
        const float  un = upot[b * NPTS + n];
        acc += __expf(-cv.x * INV_REG - un - vv.x) * cv.x;
        acc += __expf(-cv.y * INV_REG - un - vv.y) * cv.y;
        acc += __expf(-cv.z * INV_REG - un - vv.z) * cv.z;
        acc += __expf(-cv.w * INV_REG - un - vv.w) * cv.w;
    }
    sdata[tid] = acc;
    __syncthreads();
    #pragma unroll
    for (int off = 128; off >= 1; off >>= 1) {
        if (tid < off) sdata[tid] += sdata[tid + off];
        __syncthreads();
    }
    if (tid == 0) partials[blockIdx.x] = sdata[0];
}

__global__ void sk_final_reduce(const float* __restrict__ partials,
                                float* __restrict__ out) {
    __shared__ float sdata[256];
    const int tid = threadIdx.x;
    float acc = 0.0f;
    #pragma unroll
    for (int k = 0; k < 4; ++k) acc += partials[tid + k * 256];
    sdata[tid] = acc;
    __syncthreads();
    #pragma unroll
    for (int off = 128; off >= 1; off >>= 1) {
        if (tid < off) sdata[tid] += sdata[tid + off];
        __syncthreads();
    }
    // mean over batches of (sum / NUM_SAMPLES) == total / (B*N)
    if (tid == 0) out[0] = sdata[0] * (1.0f / (float)(BATCH * NPTS));
}

// ---------------------------------------------------------------------------
extern "C" void kernel_launch(void* const* d_in, const int* in_sizes, int n_in,
                              void* d_out, int out_size, void* d_ws, size_t ws_size,
                              hipStream_t stream) {
    (void)in_sizes; (void)n_in; (void)out_size; (void)ws_size;
    const float* pred = (const float*)d_in[0];
    const float* gt   = (const float*)d_in[1];

    // workspace layout: cost (16MB) | u (16KB) | v (16KB) | partials (4KB)
    float* cost     = (float*)d_ws;
    float* upot     = cost + (size_t)BATCH * NPTS * NPTS;
    float* vpot     = upot + BATCH * NPTS;
    float* partials = vpot + BATCH * NPTS;

    sk_init<<<(BATCH * NPTS + 255) / 256, 256, 0, stream>>>(vpot);

    // 16384 tiles (one wave each), 8 waves per 256-thread block
    sk_cost_wmma<<<2048, 256, 0, stream>>>(pred, gt, cost);

    for (int it = 0; it < ITERS; ++it) {
        sk_row_lse<<<512, 256, 0, stream>>>(cost, vpot, upot);  // 4096 rows
        sk_col_lse<<<32, 256, 0, stream>>>(cost, upot, vpot);   // 4096 cols, 128/block
    }

    sk_final_partial<<<1024, 256, 0, stream>>>(cost, upot, vpot, partials);
    sk_final_reduce<<<1, 256, 0, stream>>>(partials, (float*)d_out);
}